// SpikeEncoder_3238405341757
// MI455X (gfx1250) — compile-verified
//
#include <hip/hip_runtime.h>
#include <math.h>

typedef __attribute__((ext_vector_type(2))) float v2f;
typedef __attribute__((ext_vector_type(8))) float v8f;

#define SEQ_LEN_   1024
#define N_NEURONS_ 256
#define INV_NORM_  0.19947114020071635f  // 1/(2*sqrt(2*pi))

__global__ __launch_bounds__(256) void spike_zero_kernel(float4* __restrict__ out, int n4) {
  int i = blockIdx.x * 256 + threadIdx.x;
  if (i < n4) out[i] = make_float4(0.f, 0.f, 0.f, 0.f);
}

// One wave (32 lanes) handles a group of 16 events.
// Per group: C[16 events x 32 bins] = A[16x16 Taylor terms] x B[16x32]
// computed as 2 N-tiles x 4 K-steps of v_wmma_f32_16x16x4_f32,
// then scattered into the output with global_atomic_add_f32.
__global__ __launch_bounds__(256) void spike_wmma_kernel(
    const float* __restrict__ events, const int* __restrict__ batch_idx,
    float* __restrict__ out, int E) {
  const int lane = threadIdx.x & 31;
  const int wv   = threadIdx.x >> 5;
  const int h    = lane >> 4;   // half-wave: selects which K-pair / M-offset this lane holds
  const int n    = lane & 15;   // column index (bin within tile) == event index for A rows

  __shared__ float s_a[8][16];
  __shared__ float s_f[8][16];
  __shared__ int   s_row[8][16];
  __shared__ int   s_t0[8][16];

  const int  gid    = blockIdx.x * 8 + wv;   // one 16-event group per wave
  const bool active = (gid * 16) < E;

  // ---- stage per-event scalars into LDS (lanes 0-15) ----
  if (lane < 16) {
    float a = 0.f, f = 0.f;
    int row = 0, t0 = -100000;   // padding: t0 far out of range -> atomics skipped
    const int e = gid * 16 + lane;
    if (active && e < E) {
      const float te     = events[2 * e];
      const int   neuron = (int)events[2 * e + 1];
      const int   b      = batch_idx[e];
      const float t0f    = rintf(te);
      f  = te - t0f;                        // f in [-0.5, 0.5]
      t0 = (int)t0f;
      a  = INV_NORM_ * expf(-0.125f * f * f);
      row = b * N_NEURONS_ + neuron;
    }
    s_a[wv][lane] = a;  s_f[wv][lane] = f;
    s_row[wv][lane] = row;  s_t0[wv][lane] = t0;
  }
  __syncthreads();
  if (!active) return;                      // wave-uniform: EXEC all-1s for WMMA below

  // ---- B matrix: B[k, j] = exp(-j^2/8) * (j/4)^k / k!  (event-independent) ----
  // 32-bit B 4x16 layout: lanes 0-15 hold K = {0,1} in the two VGPRs, lanes 16-31 K = {2,3}.
  v2f Bm[2][4];
  #pragma unroll
  for (int tile = 0; tile < 2; ++tile) {
    const float jf = (float)(n + (tile == 0 ? -16 : 0));  // j in [-16..-1] then [0..15]
    const float w  = expf(-0.125f * jf * jf);
    const float j4 = 0.25f * jf;
    float c[16];
    c[0] = w;
    #pragma unroll
    for (int k = 1; k < 16; ++k) c[k] = c[k - 1] * (j4 * (1.0f / (float)k));
    #pragma unroll
    for (int kc = 0; kc < 4; ++kc) {
      const float b0 = h ? c[4 * kc + 2] : c[4 * kc + 0];
      const float b1 = h ? c[4 * kc + 3] : c[4 * kc + 1];
      Bm[tile][kc] = (v2f){b0, b1};
    }
  }

  // ---- A matrix: A[e, k] = a_e * f_e^k ----
  // 32-bit A 16x4 layout: lane holds M = lane&15; lanes 0-15 K={0,1}, lanes 16-31 K={2,3}.
  const float a  = s_a[wv][n];
  const float f  = s_f[wv][n];
  const float f2 = f * f;
  const float f4 = f2 * f2;
  float p = a * (h ? f2 : 1.0f);            // a * f^(2h), advanced by f^4 per K-step

  v8f C0 = {0.f,0.f,0.f,0.f,0.f,0.f,0.f,0.f};
  v8f C1 = {0.f,0.f,0.f,0.f,0.f,0.f,0.f,0.f};
  #pragma unroll
  for (int kc = 0; kc < 4; ++kc) {
    const v2f A2 = (v2f){p, p * f};
    C0 = __builtin_amdgcn_wmma_f32_16x16x4_f32(false, A2, false, Bm[0][kc],
                                               (short)0, C0, false, false);
    C1 = __builtin_amdgcn_wmma_f32_16x16x4_f32(false, A2, false, Bm[1][kc],
                                               (short)0, C1, false, false);
    p *= f4;
  }

  // ---- scatter: C element (vgpr v, lane) -> event M = v + 8h, bin j = tile_base + n ----
  #pragma unroll
  for (int v = 0; v < 8; ++v) {
    const int M   = v + 8 * h;
    const int row = s_row[wv][M];
    const int t0  = s_t0[wv][M];
    const int ta  = t0 - 16 + n;            // tile 0: j in [-16, -1]
    const int tb  = t0 + n;                 // tile 1: j in [0, 15]
    float* base = out + (size_t)row * SEQ_LEN_;
    if ((unsigned)ta < (unsigned)SEQ_LEN_) atomicAdd(base + ta, C0[v]);
    if ((unsigned)tb < (unsigned)SEQ_LEN_) atomicAdd(base + tb, C1[v]);
  }
}

extern "C" void kernel_launch(void* const* d_in, const int* in_sizes, int n_in,
                              void* d_out, int out_size, void* d_ws, size_t ws_size,
                              hipStream_t stream) {
  const float* events = (const float*)d_in[0];   // (E, 2) f32: [time, neuron]
  const int*   batch  = (const int*)d_in[1];     // (E,) i32
  float*       out    = (float*)d_out;           // (32, 256, 1024) f32
  const int E  = in_sizes[1];

  const int n4 = out_size / 4;                   // zero-fill with float4 stores
  spike_zero_kernel<<<(n4 + 255) / 256, 256, 0, stream>>>((float4*)out, n4);

  const int groups = (E + 15) / 16;              // 16 events per wave
  const int blocks = (groups + 7) / 8;           // 8 waves per 256-thread block
  spike_wmma_kernel<<<blocks, 256, 0, stream>>>(events, batch, out, E);
}